// attention_1065151889732
// MI455X (gfx1250) — compile-verified
//
#include <hip/hip_runtime.h>
#include <stdint.h>

// ---------------------------------------------------------------------------
// out = (q_in@wQ) @ ((k_in@wK)^T) / 64 @ (v_in@wV)
// Re-associated (no softmax in reference -> matmul chain is associative):
//   P  = k_in^T @ v_in                 [256 x 256]   (split-N partials + reduce)
//   T1 = P @ wV                        [256 x 64]
//   M  = (wK^T @ T1) / 64              [64  x 64]
//   W2 = wQ @ M                        [256 x 64]
//   out= q_in @ W2                     [8192 x 64]
// ~1.1 GFLOP, ~26 MB mandatory HBM traffic -> memory bound (~1.1us @ 23.3TB/s).
// fp32 throughout via V_WMMA_F32_16X16X4_F32 (wave32, 16x16 tile per wave).
// ---------------------------------------------------------------------------

typedef float v2f __attribute__((ext_vector_type(2)));
typedef float v8f __attribute__((ext_vector_type(8)));

#define WMMA_F32(a, b, c) \
    __builtin_amdgcn_wmma_f32_16x16x4_f32(false, (a), false, (b), (short)0, (c), false, false)

static constexpr int N_ROWS = 8192;
static constexpr int D_IN   = 256;
static constexpr int DK     = 64;
static constexpr int NSLICE = 8;     // split-N factor for P partials

// --- gfx1250 async global->LDS staging path --------------------------------
#if defined(__has_builtin)
#  if __has_builtin(__builtin_amdgcn_global_load_async_to_lds_b128) && \
      __has_builtin(__builtin_amdgcn_s_wait_asynccnt)
#    define USE_ASYNC_LDS 1
#  endif
#endif

// Builtin prototype (from hipcc diagnostic): param0 is generic-vector int4 in
// HIP __device__ (addrspace(1)); param1 is the LDS (addrspace(3)) side.
typedef int v4i __attribute__((vector_size(16)));
typedef __attribute__((address_space(1))) v4i glb_v4i;
typedef __attribute__((address_space(3))) v4i lds_v4i;

__device__ __forceinline__ void stage16(float* lds_dst, const float* g_src) {
#ifdef USE_ASYNC_LDS
    // int->pointer casts: global generic == as(1) value; LDS offset is the low
    // 32 bits of the generic aperture address, so truncation is correct.
    __builtin_amdgcn_global_load_async_to_lds_b128(
        (glb_v4i*)(uintptr_t)g_src, (lds_v4i*)(uintptr_t)lds_dst, 0, 0);
#else
    *(float4*)lds_dst = *(const float4*)g_src;
#endif
}
__device__ __forceinline__ void stage_join() {
#ifdef USE_ASYNC_LDS
    __builtin_amdgcn_s_wait_asynccnt(0);
#endif
}

// D-matrix layout (f32 16x16): VGPR r, lanes 0-15 -> row r, lanes 16-31 -> row r+8.
__device__ __forceinline__ void store_tile(float* __restrict__ C, int ldc,
                                           int row0, int col0, v8f c, int lane,
                                           float scale = 1.0f) {
    const int col   = lane & 15;
    const int rbase = (lane >> 4) * 8;
#pragma unroll
    for (int r = 0; r < 8; ++r)
        C[(row0 + rbase + r) * ldc + col0 + col] = c[r] * scale;
}

// One 16x16 output tile of row-major A[lda] @ row-major B[ldb], K steps of 4.
__device__ __forceinline__ v8f gemm_tile(const float* __restrict__ A, int lda,
                                         const float* __restrict__ B, int ldb,
                                         int row0, int col0, int K, int lane) {
    const int m    = lane & 15;
    const int koff = (lane >> 4) << 1;  // 0 or 2
    const float* ap = A + (row0 + m) * lda + koff;
    const float* bp = B + koff * ldb + col0 + m;
    v8f acc = {0.f, 0.f, 0.f, 0.f, 0.f, 0.f, 0.f, 0.f};
#pragma unroll 4
    for (int k = 0; k < K; k += 4) {
        v2f a = *(const v2f*)(ap + k);
        v2f b;
        b.x = bp[k * ldb];
        b.y = bp[(k + 1) * ldb];
        acc = WMMA_F32(a, b, acc);
    }
    return acc;
}

// ---------------------------------------------------------------------------
// Kernel 1: Ppart[slice] = k_in^T @ v_in over slice's 1024 rows.
// Grid: 128 blocks = 8 N-slices x 16 (4x4) 64x64 P-tiles.
// Stage k/v 64x64 chunks into LDS with async b128 copies, then 8 waves x 2
// disjoint 16x16 sub-tiles each -> no cross-wave reduction, no atomics.
// LDS row stride 80 (mod 64 banks = 16): half-wave K-split lands on disjoint
// bank ranges -> conflict-free ds reads.
// ---------------------------------------------------------------------------
__global__ __launch_bounds__(256) void ktv_part_kernel(const float* __restrict__ k_in,
                                                       const float* __restrict__ v_in,
                                                       float* __restrict__ Ppart) {
    constexpr int LDB = 80;
    __shared__ float kC[64 * LDB];
    __shared__ float vC[64 * LDB];

    const int lane  = threadIdx.x & 31;
    const int wave  = threadIdx.x >> 5;
    const int slice = blockIdx.x >> 4;       // 0..7
    const int tile  = blockIdx.x & 15;       // 0..15 (4x4 of 64x64)
    const int tI    = (tile >> 2) * 64;      // P row block  (k columns)
    const int tJ    = (tile & 3) * 64;       // P col block  (v columns)
    const int m     = lane & 15;
    const int koff  = (lane >> 4) << 1;

    const int st0 = wave * 2, st1 = st0 + 1;         // sub-tiles (4x4 grid)
    const int si0 = (st0 >> 2) * 16, sj0 = (st0 & 3) * 16;
    const int si1 = (st1 >> 2) * 16, sj1 = (st1 & 3) * 16;

    v8f acc0 = {0.f,0.f,0.f,0.f,0.f,0.f,0.f,0.f};
    v8f acc1 = acc0;

    const int n0 = slice * (N_ROWS / NSLICE);        // 1024 rows per slice
    const int sr = threadIdx.x >> 4;                 // staging row 0..15
    const int sc = (threadIdx.x & 15) * 4;           // staging col 0..60

    for (int c = 0; c < 16; ++c) {
        const int nb = n0 + c * 64;
#pragma unroll
        for (int rr = 0; rr < 4; ++rr) {
            const int r = sr + rr * 16;
            stage16(&kC[r * LDB + sc], &k_in[(nb + r) * D_IN + tI + sc]);
            stage16(&vC[r * LDB + sc], &v_in[(nb + r) * D_IN + tJ + sc]);
        }
        stage_join();
        __syncthreads();
#pragma unroll 4
        for (int kk = 0; kk < 64; kk += 4) {
            const int kr = kk + koff;
            v2f a, b;
            a.x = kC[kr * LDB + si0 + m];  a.y = kC[(kr + 1) * LDB + si0 + m];
            b.x = vC[kr * LDB + sj0 + m];  b.y = vC[(kr + 1) * LDB + sj0 + m];
            acc0 = WMMA_F32(a, b, acc0);
            a.x = kC[kr * LDB + si1 + m];  a.y = kC[(kr + 1) * LDB + si1 + m];
            b.x = vC[kr * LDB + sj1 + m];  b.y = vC[(kr + 1) * LDB + sj1 + m];
            acc1 = WMMA_F32(a, b, acc1);
        }
        __syncthreads();
    }

    float* out = Ppart + slice * (D_IN * D_IN);
    store_tile(out, D_IN, tI + si0, tJ + sj0, acc0, lane);
    store_tile(out, D_IN, tI + si1, tJ + sj1, acc1, lane);
}

// ---------------------------------------------------------------------------
// Kernel 2: P = sum over 8 slices of Ppart.  65536 floats, float4 per thread.
// ---------------------------------------------------------------------------
__global__ __launch_bounds__(256) void reduce_p_kernel(const float* __restrict__ Ppart,
                                                       float* __restrict__ P) {
    const int i = (blockIdx.x * 256 + threadIdx.x) * 4;
    float4 s = *(const float4*)&Ppart[i];
#pragma unroll
    for (int sl = 1; sl < NSLICE; ++sl) {
        float4 t = *(const float4*)&Ppart[sl * (D_IN * D_IN) + i];
        s.x += t.x; s.y += t.y; s.z += t.z; s.w += t.w;
    }
    *(float4*)&P[i] = s;
}

// ---------------------------------------------------------------------------
// Kernel 3: T1 = P @ wV   [256x64]. 64 tiles, 8 blocks x 8 waves.
// ---------------------------------------------------------------------------
__global__ __launch_bounds__(256) void pv_kernel(const float* __restrict__ P,
                                                 const float* __restrict__ wV,
                                                 float* __restrict__ T1) {
    const int lane   = threadIdx.x & 31;
    const int waveId = blockIdx.x * 8 + (threadIdx.x >> 5);
    const int tileI  = waveId >> 2;
    const int tileJ  = waveId & 3;
    v8f c = gemm_tile(P, D_IN, wV, DK, tileI * 16, tileJ * 16, D_IN, lane);
    store_tile(T1, DK, tileI * 16, tileJ * 16, c, lane);
}

// ---------------------------------------------------------------------------
// Kernel 4: M = (wK^T @ T1) / 64   [64x64]. 16 tiles, 1 block, 2 tiles/wave.
// ---------------------------------------------------------------------------
__global__ __launch_bounds__(256) void ktt1_kernel(const float* __restrict__ wK,
                                                   const float* __restrict__ T1,
                                                   float* __restrict__ M) {
    const int lane = threadIdx.x & 31;
    const int wave = threadIdx.x >> 5;
    const int m    = lane & 15;
    const int koff = (lane >> 4) << 1;
#pragma unroll
    for (int i = 0; i < 2; ++i) {
        const int t     = wave * 2 + i;
        const int tileI = t >> 2;
        const int tileJ = t & 3;
        const float* ka = wK + tileI * 16 + m;   // A[m][k] = wK[k][tileI*16+m]
        const float* bp = T1 + tileJ * 16 + m;
        v8f acc = {0.f,0.f,0.f,0.f,0.f,0.f,0.f,0.f};
#pragma unroll 4
        for (int k = 0; k < D_IN; k += 4) {
            v2f a, b;
            a.x = ka[(k + koff) * DK];
            a.y = ka[(k + koff + 1) * DK];
            b.x = bp[(k + koff) * DK];
            b.y = bp[(k + koff + 1) * DK];
            acc = WMMA_F32(a, b, acc);
        }
        store_tile(M, DK, tileI * 16, tileJ * 16, acc, lane, 1.0f / 64.0f);
    }
}

// ---------------------------------------------------------------------------
// Kernel 5: W2 = wQ @ M   [256x64]. 64 tiles, 8 blocks x 8 waves.
// ---------------------------------------------------------------------------
__global__ __launch_bounds__(256) void wqm_kernel(const float* __restrict__ wQ,
                                                  const float* __restrict__ M,
                                                  float* __restrict__ W2) {
    const int lane   = threadIdx.x & 31;
    const int waveId = blockIdx.x * 8 + (threadIdx.x >> 5);
    const int tileI  = waveId >> 2;
    const int tileJ  = waveId & 3;
    v8f c = gemm_tile(wQ, DK, M, DK, tileI * 16, tileJ * 16, DK, lane);
    store_tile(W2, DK, tileI * 16, tileJ * 16, c, lane);
}

// ---------------------------------------------------------------------------
// Kernel 6: out = q_in @ W2   [8192x64]. One 16x64 row stripe per wave
// (4 C-tiles) so q_in is streamed exactly once. W2 (64 KB) staged in LDS.
// ---------------------------------------------------------------------------
__global__ __launch_bounds__(256) void qw2_kernel(const float* __restrict__ q_in,
                                                  const float* __restrict__ W2,
                                                  float* __restrict__ out) {
    __shared__ float w2s[D_IN * DK];  // 64 KB
    for (int i = threadIdx.x * 4; i < D_IN * DK; i += 256 * 4)
        *(float4*)&w2s[i] = *(const float4*)&W2[i];
    __syncthreads();

    const int lane = threadIdx.x & 31;
    const int wave = threadIdx.x >> 5;
    const int row0 = (blockIdx.x * 8 + wave) * 16;
    const int m    = lane & 15;
    const int koff = (lane >> 4) << 1;

    const float* ap = q_in + (row0 + m) * D_IN + koff;
    v8f c0 = {0.f,0.f,0.f,0.f,0.f,0.f,0.f,0.f};
    v8f c1 = c0, c2 = c0, c3 = c0;
#pragma unroll 4
    for (int k = 0; k < D_IN; k += 4) {
        v2f a = *(const v2f*)(ap + k);
        const float* b0 = &w2s[(k + koff) * DK + m];
        const float* b1 = &w2s[(k + koff + 1) * DK + m];
        v2f b;
        b.x = b0[0];  b.y = b1[0];  c0 = WMMA_F32(a, b, c0);
        b.x = b0[16]; b.y = b1[16]; c1 = WMMA_F32(a, b, c1);
        b.x = b0[32]; b.y = b1[32]; c2 = WMMA_F32(a, b, c2);
        b.x = b0[48]; b.y = b1[48]; c3 = WMMA_F32(a, b, c3);
    }
    store_tile(out, DK, row0,  0, c0, lane);
    store_tile(out, DK, row0, 16, c1, lane);
    store_tile(out, DK, row0, 32, c2, lane);
    store_tile(out, DK, row0, 48, c3, lane);
}

// ---------------------------------------------------------------------------
extern "C" void kernel_launch(void* const* d_in, const int* in_sizes, int n_in,
                              void* d_out, int out_size, void* d_ws, size_t ws_size,
                              hipStream_t stream) {
    const float* q_in = (const float*)d_in[0];
    const float* k_in = (const float*)d_in[1];
    const float* v_in = (const float*)d_in[2];
    const float* wQ   = (const float*)d_in[3];
    const float* wK   = (const float*)d_in[4];
    const float* wV   = (const float*)d_in[5];
    float* out = (float*)d_out;

    // ws (floats): Ppart[8*65536] P[65536] T1[16384] M[4096] W2[16384] ~ 2.4 MB
    float* Ppart = (float*)d_ws;
    float* P     = Ppart + NSLICE * D_IN * D_IN;
    float* T1    = P + D_IN * D_IN;
    float* M     = T1 + D_IN * DK;
    float* W2    = M + DK * DK;

    hipLaunchKernelGGL(ktv_part_kernel, dim3(128), dim3(256), 0, stream, k_in, v_in, Ppart);
    hipLaunchKernelGGL(reduce_p_kernel, dim3(64),  dim3(256), 0, stream, Ppart, P);
    hipLaunchKernelGGL(pv_kernel,       dim3(8),   dim3(256), 0, stream, P, wV, T1);
    hipLaunchKernelGGL(ktt1_kernel,     dim3(1),   dim3(256), 0, stream, wK, T1, M);
    hipLaunchKernelGGL(wqm_kernel,      dim3(8),   dim3(256), 0, stream, wQ, M, W2);
    hipLaunchKernelGGL(qw2_kernel,      dim3(64),  dim3(256), 0, stream, q_in, W2, out);
}